// GuidedFilter_50723563765881
// MI455X (gfx1250) — compile-verified
//
#include <hip/hip_runtime.h>

// Guided filter, fully fused, MI455X (gfx1250, wave32).
// One wave per 16x16 output tile.
//  - raw halo tiles moved Global->LDS with GLOBAL_LOAD_ASYNC_TO_LDS_B128
//    (ASYNCcnt pipelined: p-channel loads overlap earlier compute)
//  - stage-1 7x7 box via running sums (VALU+LDS)
//  - stage-2 7x7 box via V_WMMA_F32_16X16X4_F32 banded matrix products

typedef __attribute__((ext_vector_type(2))) float v2f;
typedef __attribute__((ext_vector_type(8))) float v8f;

#define HW   1024
#define EPSF 1e-6f

// Async-copy a 32x32 f32 tile (row stride HW in global, 33 in LDS) into LDS.
// 8 x global_load_async_to_lds_b128: lane -> (row = i*4 + lane/8, 4 cols).
__device__ __forceinline__ void async_tile_load(const float* __restrict__ g,
                                                unsigned ldsbase, int lane)
{
    const int row0 = lane >> 3;          // 0..3
    const int colg = (lane & 7) * 4;     // 0,4,...,28
    #pragma unroll
    for (int i = 0; i < 8; ++i) {
        const int row  = i * 4 + row0;
        const unsigned goff = (unsigned)((row * HW + colg) * 4);
        const unsigned loff = ldsbase + (unsigned)((row * 33 + colg) * 4);
        asm volatile("global_load_async_to_lds_b128 %0, %1, %2"
                     :: "v"(loff), "v"(goff), "s"((unsigned long long)(uintptr_t)g)
                     : "memory");
    }
}

// Scalar (bounds-checked, zero-padded) tile fill for edge blocks.
__device__ __forceinline__ void scalar_tile_load(const float* __restrict__ base,
                                                 float dst[32][33],
                                                 int oy, int ox, int lane)
{
    #pragma unroll 4
    for (int r = 0; r < 32; ++r) {
        const int gy = oy - 9 + r;
        const int gx = ox - 9 + lane;
        float v = 0.0f;
        if (gy >= 0 && gy < HW && gx >= 0 && gx < HW) v = base[(size_t)gy * HW + gx];
        dst[r][lane] = v;
    }
}

// Stage-2 separable 7-tap box as S(16x22) * F(22x22) * S^T(22x16) using
// f32 WMMA 16x16x4, K padded to 24. F must be zero outside [0..21]x[0..21].
// Result is the UNNORMALIZED 49x box sum in D-layout (v8f per lane).
__device__ __forceinline__ v8f box2_wmma(const float F[24][32], float (*T)[33], int lane)
{
    const int m    = lane & 15;
    const int hi   = lane >> 4;         // 0 or 1
    const int kofs = hi ? 2 : 0;

    // Pass 1: T[y][v] = sum_{k=y..y+6} F[k][v]  (A = band, B = data)
    #pragma unroll
    for (int t = 0; t < 2; ++t) {
        const int vbase = t * 16;
        v8f acc = {};
        #pragma unroll
        for (int c = 0; c < 6; ++c) {
            const int k0 = 4 * c + kofs;
            v2f a, b;
            a[0] = (k0     >= m && k0     <= m + 6) ? 1.0f : 0.0f;
            a[1] = (k0 + 1 >= m && k0 + 1 <= m + 6) ? 1.0f : 0.0f;
            b[0] = F[k0][vbase + m];
            b[1] = F[k0 + 1][vbase + m];
            acc = __builtin_amdgcn_wmma_f32_16x16x4_f32(false, a, false, b,
                                                        (short)0, acc, false, false);
        }
        #pragma unroll
        for (int i = 0; i < 8; ++i)
            T[i + 8 * hi][vbase + m] = acc[i];
    }

    // Pass 2: R[y][x] = sum_{v=x..x+6} T[y][v]  (A = data, B = band)
    v8f acc = {};
    #pragma unroll
    for (int c = 0; c < 6; ++c) {
        const int k0 = 4 * c + kofs;
        v2f a, b;
        a[0] = T[m][k0];
        a[1] = T[m][k0 + 1];
        b[0] = (k0     >= m && k0     <= m + 6) ? 1.0f : 0.0f;
        b[1] = (k0 + 1 >= m && k0 + 1 <= m + 6) ? 1.0f : 0.0f;
        acc = __builtin_amdgcn_wmma_f32_16x16x4_f32(false, a, false, b,
                                                    (short)0, acc, false, false);
    }
    return acc;
}

__global__ __launch_bounds__(32) void guided_filter_kernel(
    const float* __restrict__ I, const float* __restrict__ P,
    float* __restrict__ Q)
{
    // Raw halo region: 32x32 covering [oy-9 .. oy+22] x [ox-9 .. ox+22]
    __shared__ float rawI[32][33];
    __shared__ float rawP[3][32][33];
    __shared__ float h1[32][33];     // hbox scratch; reused as T[16][33] in stage 2
    __shared__ float mI[22][23];
    __shared__ float iV[22][23];
    __shared__ float mP[22][23];
    __shared__ float Aab[24][32];    // 'a' field, zero-padded for WMMA K/N tiles
    __shared__ float Bab[24][32];    // 'b' field

    const int   lane  = threadIdx.x;
    const int   ox    = blockIdx.x * 16;
    const int   oy    = blockIdx.y * 16;
    const int   n     = blockIdx.z;
    const float inv49 = 1.0f / 49.0f;

    const float* Ibase = I + (size_t)n * HW * HW;
    const bool interior = (ox >= 9) && (ox + 22 < HW) && (oy >= 9) && (oy + 22 < HW);

    // ---- issue all raw tile loads (async for interior, scalar for edge) ----
    if (interior) {
        const size_t corner = (size_t)(oy - 9) * HW + (ox - 9);
        async_tile_load(Ibase + corner, (unsigned)(uintptr_t)&rawI[0][0], lane);
        #pragma unroll
        for (int c = 0; c < 3; ++c) {
            const float* Pb = P + ((size_t)(n * 3 + c)) * HW * HW + corner;
            async_tile_load(Pb, (unsigned)(uintptr_t)&rawP[c][0][0], lane);
        }
    } else {
        scalar_tile_load(Ibase, rawI, oy, ox, lane);
        #pragma unroll
        for (int c = 0; c < 3; ++c)
            scalar_tile_load(P + ((size_t)(n * 3 + c)) * HW * HW, rawP[c], oy, ox, lane);
    }

    // ---- zero WMMA padding of a/b once (pad region never rewritten) ----
    for (int idx = lane; idx < 24 * 32; idx += 32) {
        (&Aab[0][0])[idx] = 0.0f;
        (&Bab[0][0])[idx] = 0.0f;
    }

    // I tile ready (first 8 async ops; async loads complete in order).
    asm volatile("s_wait_asynccnt 0x18" ::: "memory");

    // ---- stage-1 hbox of I, vbox -> mean_I ----
    {
        const int row = lane;
        float s = 0.0f;
        #pragma unroll
        for (int c = 0; c < 7; ++c) s += rawI[row][c];
        h1[row][3] = s;
        for (int j = 4; j <= 28; ++j) { s += rawI[row][j + 3] - rawI[row][j - 4]; h1[row][j] = s; }
    }
    for (int idx = lane; idx < 484; idx += 32) {
        const int u = idx / 22, v = idx - u * 22;
        float s = 0.0f;
        #pragma unroll
        for (int d = 0; d < 7; ++d) s += h1[u + 3 + d][v + 6];
        mI[u][v] = s * inv49;
    }

    // ---- stage-1 hbox of I*I, vbox -> 1/(var+eps) ----
    {
        const int row = lane;
        float s = 0.0f;
        #pragma unroll
        for (int c = 0; c < 7; ++c) { const float t = rawI[row][c]; s += t * t; }
        h1[row][3] = s;
        for (int j = 4; j <= 28; ++j) {
            const float tp = rawI[row][j + 3], tm = rawI[row][j - 4];
            s += tp * tp - tm * tm;
            h1[row][j] = s;
        }
    }
    for (int idx = lane; idx < 484; idx += 32) {
        const int u = idx / 22, v = idx - u * 22;
        float s = 0.0f;
        #pragma unroll
        for (int d = 0; d < 7; ++d) s += h1[u + 3 + d][v + 6];
        const float mi = mI[u][v];
        iV[u][v] = 1.0f / (s * inv49 - mi * mi + EPSF);
    }

    // ---- channels ----
    for (int ch = 0; ch < 3; ++ch) {
        // p[ch] tile ready: waits exploit in-order async completion; NOP on edge path.
        if (ch == 0)      asm volatile("s_wait_asynccnt 0x10" ::: "memory");
        else if (ch == 1) asm volatile("s_wait_asynccnt 0x8"  ::: "memory");
        else              asm volatile("s_wait_asynccnt 0x0"  ::: "memory");

        const float (*rp)[33] = rawP[ch];

        // hbox(p) -> h1 ; vbox -> mean_p
        {
            const int row = lane;
            float s = 0.0f;
            #pragma unroll
            for (int c = 0; c < 7; ++c) s += rp[row][c];
            h1[row][3] = s;
            for (int j = 4; j <= 28; ++j) { s += rp[row][j + 3] - rp[row][j - 4]; h1[row][j] = s; }
        }
        for (int idx = lane; idx < 484; idx += 32) {
            const int u = idx / 22, v = idx - u * 22;
            float s = 0.0f;
            #pragma unroll
            for (int d = 0; d < 7; ++d) s += h1[u + 3 + d][v + 6];
            mP[u][v] = s * inv49;
        }

        // hbox(I*p) -> h1 ; vbox -> mean_Ip, then a,b (zeroed outside image)
        {
            const int row = lane;
            float s = 0.0f;
            #pragma unroll
            for (int c = 0; c < 7; ++c) s += rawI[row][c] * rp[row][c];
            h1[row][3] = s;
            for (int j = 4; j <= 28; ++j) {
                s += rawI[row][j + 3] * rp[row][j + 3] - rawI[row][j - 4] * rp[row][j - 4];
                h1[row][j] = s;
            }
        }
        for (int idx = lane; idx < 484; idx += 32) {
            const int u = idx / 22, v = idx - u * 22;
            float s = 0.0f;
            #pragma unroll
            for (int d = 0; d < 7; ++d) s += h1[u + 3 + d][v + 6];
            const float mip = s * inv49;
            const float mi = mI[u][v], mp = mP[u][v];
            const float a = (mip - mi * mp) * iV[u][v];
            const float b = mp - a * mi;
            const int ay = oy - 3 + u, ax = ox - 3 + v;
            const bool ok = (ay >= 0) && (ay < HW) && (ax >= 0) && (ax < HW);
            Aab[u][v] = ok ? a : 0.0f;
            Bab[u][v] = ok ? b : 0.0f;
        }

        // ---- stage-2 box via WMMA (matrix unit) ----
        float (*T)[33] = h1;
        const v8f rA = box2_wmma(Aab, T, lane);
        const v8f rB = box2_wmma(Bab, T, lane);

        // ---- q = clip(mean_a * I + mean_b) and store (D-layout) ----
        float* Qbase = Q + ((size_t)(n * 3 + ch)) * HW * HW;
        const int m = lane & 15, hi = lane >> 4;
        #pragma unroll
        for (int i = 0; i < 8; ++i) {
            const int y = i + 8 * hi;
            const float Ival = rawI[y + 9][m + 9];
            float q = rA[i] * inv49 * Ival + rB[i] * inv49;
            q = fminf(fmaxf(q, 0.0f), 1.0f);
            Qbase[(size_t)(oy + y) * HW + (ox + m)] = q;
        }
    }
}

extern "C" void kernel_launch(void* const* d_in, const int* in_sizes, int n_in,
                              void* d_out, int out_size, void* d_ws, size_t ws_size,
                              hipStream_t stream) {
    const float* I = (const float*)d_in[0];   // (B,1,1024,1024) f32
    const float* P = (const float*)d_in[1];   // (B,3,1024,1024) f32
    float* Q = (float*)d_out;                 // (B,3,1024,1024) f32
    const int B = in_sizes[0] / (HW * HW);    // batch from guide size
    dim3 grid(HW / 16, HW / 16, B);
    guided_filter_kernel<<<grid, dim3(32, 1, 1), 0, stream>>>(I, P, Q);
}